// FilterDetections_17463337026045
// MI455X (gfx1250) — compile-verified
//
#include <hip/hip_runtime.h>
#include <hip/hip_bf16.h>
#include <stdint.h>

// Problem constants (match reference)
#define NMS_THRESH   0.5f
#define SCORE_THRESH 0.05f
#define MAX_DET      300
#define BB           8
#define NN           20000
#define CC           80
#define PP           51

#define THREADS 1024
#define SLOTS1  20            // ceil(NN / THREADS)
#define CM      (CC * MAX_DET) // 24000 candidates per batch
#define SLOTS2  24            // ceil(CM / THREADS)

typedef __attribute__((address_space(3))) char lds_char_t;

__device__ __forceinline__ unsigned lds_byte_addr(void* p) {
  // generic -> LDS(addrspace 3) cast; AS3 pointers are 32-bit LDS offsets
  return (unsigned)(size_t)(lds_char_t*)p;
}

__device__ __forceinline__ float neg_inf() { return __int_as_float(0xff800000); }

// wave32 argmax: higher score wins, tie -> lower index (matches jnp.argmax)
__device__ __forceinline__ void argmax32(float& s, int& i) {
#pragma unroll
  for (int off = 16; off > 0; off >>= 1) {
    float os = __shfl_xor(s, off, 32);
    int   oi = __shfl_xor(i, off, 32);
    if (os > s || (os == s && oi < i)) { s = os; i = oi; }
  }
}

// ---------------------------------------------------------------------------
// Kernel 1: per-(batch,class) greedy NMS. One 1024-thread workgroup per pair.
// Boxes for the batch are async-staged into LDS (312.5 KB of the 320 KB WGP
// LDS) while per-class scores stream into VGPRs.
// ---------------------------------------------------------------------------
__global__ __launch_bounds__(THREADS)
void fd_nms_per_class(const float* __restrict__ boxes,
                      const float* __restrict__ cls,
                      float* __restrict__ cand_s,
                      int*   __restrict__ cand_i) {
  extern __shared__ unsigned char smem[];
  float4* sbox = (float4*)smem;                                   // NN float4
  float*  rs   = (float*)(smem + (size_t)NN * 16);                // 32 floats
  int*    ri   = (int*)  (smem + (size_t)NN * 16 + 128);          // 32 ints
  float*  bcf  = (float*)(smem + (size_t)NN * 16 + 256);          // broadcast
  int*    bci  = (int*)bcf;

  const int t  = threadIdx.x;
  const int bc = blockIdx.x;     // b*CC + c
  const int b  = bc / CC;
  const int c  = bc % CC;

  // --- async stage boxes[b] -> LDS (ASYNCcnt path, overlaps score loads) ---
  const float4* gbox = (const float4*)(boxes + (size_t)b * NN * 4);
  for (int i = t; i < NN; i += THREADS) {
    unsigned           lo = lds_byte_addr(&sbox[i]);
    unsigned long long ga = (unsigned long long)(uintptr_t)(&gbox[i]);
    asm volatile("global_load_async_to_lds_b128 %0, %1, off"
                 :: "v"(lo), "v"(ga) : "memory");
  }

  // --- load + threshold-mask this class's scores into VGPRs ---
  float sc[SLOTS1];
  const float* gcls = cls + (size_t)b * NN * CC + c;
#pragma unroll
  for (int k = 0; k < SLOTS1; ++k) {
    int idx = k * THREADS + t;
    float v = neg_inf();
    if (idx < NN) {
      float x = gcls[(size_t)idx * CC];
      v = (x > SCORE_THRESH) ? x : neg_inf();
    }
    sc[k] = v;
  }

  asm volatile("s_wait_asynccnt 0x0" ::: "memory");
  __syncthreads();

  float* out_s = cand_s + (size_t)bc * MAX_DET;
  int*   out_i = cand_i + (size_t)bc * MAX_DET;

  for (int it = 0; it < MAX_DET; ++it) {
    // local argmax over this thread's 20 scores
    float bs = neg_inf(); int bi = 0x7fffffff;
#pragma unroll
    for (int k = 0; k < SLOTS1; ++k) {
      int idx = k * THREADS + t;
      if (sc[k] > bs) { bs = sc[k]; bi = idx; }
    }
    argmax32(bs, bi);
    if ((t & 31) == 0) { rs[t >> 5] = bs; ri[t >> 5] = bi; }
    __syncthreads();
    if (t < 32) {
      float s2 = rs[t]; int i2 = ri[t];
      argmax32(s2, i2);
      if (t == 0) {
        bcf[0] = s2; bci[1] = i2;
        if (s2 > neg_inf()) {
          float4 bb = sbox[i2];
          bcf[2] = bb.x; bcf[3] = bb.y; bcf[4] = bb.z; bcf[5] = bb.w;
          out_s[it] = s2; out_i[it] = i2;
        } else {
          out_s[it] = neg_inf(); out_i[it] = 0;
        }
      }
    }
    __syncthreads();

    float ws = bcf[0];
    if (ws > neg_inf()) {
      int   wi = bci[1];
      float x1 = bcf[2], y1 = bcf[3], x2 = bcf[4], y2 = bcf[5];
      float areaB = (x2 - x1) * (y2 - y1);
      // divide-free suppression: iou > T  <=>  inter > T*union (union > 0)
#pragma unroll
      for (int k = 0; k < SLOTS1; ++k) {
        int idx = k * THREADS + t;
        if (sc[k] > neg_inf()) {
          float4 nb  = sbox[idx];                       // ds_load_b128
          float ix1 = fmaxf(x1, nb.x), iy1 = fmaxf(y1, nb.y);
          float ix2 = fminf(x2, nb.z), iy2 = fminf(y2, nb.w);
          float inter = fmaxf(ix2 - ix1, 0.f) * fmaxf(iy2 - iy1, 0.f);
          float areaN = (nb.z - nb.x) * (nb.w - nb.y);
          float uni   = areaB + areaN - inter;
          bool  sup   = (uni > 0.f) && (inter > NMS_THRESH * uni);
          if (sup || idx == wi) sc[k] = neg_inf();
        }
      }
    }
    // no trailing barrier needed: bcf is rewritten only after the next
    // iteration's first __syncthreads()
  }
}

// ---------------------------------------------------------------------------
// Kernel 2: per-batch global top-300 over 80*300 candidates + output writes.
// Candidate scores (96 KB) async-staged into LDS; 300 iterative argmax picks.
// ---------------------------------------------------------------------------
__global__ __launch_bounds__(THREADS)
void fd_topk_finalize(const float* __restrict__ boxes,
                      const float* __restrict__ rel,
                      const float* __restrict__ cand_s,
                      const int*   __restrict__ cand_i,
                      float* __restrict__ out_boxes,
                      float* __restrict__ out_scores,
                      int*   __restrict__ out_labels,
                      float* __restrict__ pred_scores,
                      int*   __restrict__ pred_labels) {
  extern __shared__ unsigned char smem[];
  float* s   = (float*)smem;                          // CM floats
  float* rs  = (float*)(smem + (size_t)CM * 4);       // 32 floats
  int*   ri  = (int*)  (smem + (size_t)CM * 4 + 128); // 32 ints
  float* bcf = (float*)(smem + (size_t)CM * 4 + 256); // broadcast
  int*   bci = (int*)bcf;

  const int t = threadIdx.x;
  const int b = blockIdx.x;

  const float* gs = cand_s + (size_t)b * CM;
  for (int p = t; p < CM; p += THREADS) {
    unsigned           lo = lds_byte_addr(&s[p]);
    unsigned long long ga = (unsigned long long)(uintptr_t)(&gs[p]);
    asm volatile("global_load_async_to_lds_b32 %0, %1, off"
                 :: "v"(lo), "v"(ga) : "memory");
  }
  asm volatile("s_wait_asynccnt 0x0" ::: "memory");
  __syncthreads();

  for (int j = 0; j < MAX_DET; ++j) {
    float bsv = neg_inf(); int bp = 0x7fffffff;
#pragma unroll
    for (int k = 0; k < SLOTS2; ++k) {
      int p = k * THREADS + t;
      if (p < CM) { float v = s[p]; if (v > bsv) { bsv = v; bp = p; } }
    }
    argmax32(bsv, bp);
    if ((t & 31) == 0) { rs[t >> 5] = bsv; ri[t >> 5] = bp; }
    __syncthreads();
    if (t < 32) {
      float s2 = rs[t]; int p2 = ri[t];
      argmax32(s2, p2);
      if (t == 0) {
        bcf[0] = s2; bci[1] = p2;
        if (s2 > neg_inf()) s[p2] = neg_inf();   // consume this candidate
      }
    }
    __syncthreads();

    float ts = bcf[0]; int tp = bci[1];
    bool valid = ts > neg_inf();

    if (t < 32) {   // wave 0 does the gathers + 51-wide relationship argmax
      int label = -1, bi = 0, pl = -1;
      float ps = -1.f;
      float4 bb = make_float4(-1.f, -1.f, -1.f, -1.f);
      if (valid) {
        label = tp / MAX_DET;                       // class id (flatten (C,M))
        bi = cand_i[(size_t)b * CM + tp];
        const float* rr = rel + ((size_t)b * NN + bi) * PP;
        float m = neg_inf(); int ml = 0x7fffffff;
        for (int l = t; l < PP; l += 32) {
          float v = rr[l];
          if (v > m) { m = v; ml = l; }
        }
        argmax32(m, ml);
        ps = m; pl = ml;
        if (t == 0) bb = ((const float4*)boxes)[(size_t)b * NN + bi];
      }
      if (t == 0) {
        int o = b * MAX_DET + j;
        ((float4*)out_boxes)[o] = bb;
        out_scores[o]  = valid ? ts : -1.f;
        out_labels[o]  = valid ? label : -1;
        pred_scores[o] = valid ? ps : -1.f;
        pred_labels[o] = valid ? pl : -1;
      }
    }
    __syncthreads();
  }
}

// ---------------------------------------------------------------------------
extern "C" void kernel_launch(void* const* d_in, const int* in_sizes, int n_in,
                              void* d_out, int out_size, void* d_ws, size_t ws_size,
                              hipStream_t stream) {
  (void)in_sizes; (void)n_in; (void)out_size; (void)ws_size;

  const float* boxes = (const float*)d_in[0];  // (B,N,4)
  const float* cls   = (const float*)d_in[1];  // (B,N,C)
  const float* rel   = (const float*)d_in[2];  // (B,N,P)

  // workspace: candidate lists, (B,C,MAX_DET) scores + indices (~1.5 MB)
  float* cand_s = (float*)d_ws;
  int*   cand_i = (int*)(cand_s + (size_t)BB * CC * MAX_DET);

  // outputs concatenated flat in reference return order
  float* out_boxes   = (float*)d_out;                              // B*300*4
  float* out_scores  = out_boxes + (size_t)BB * MAX_DET * 4;       // B*300
  int*   out_labels  = (int*)(out_scores + (size_t)BB * MAX_DET);  // B*300
  float* pred_scores = (float*)(out_labels + (size_t)BB * MAX_DET);// B*300
  int*   pred_labels = (int*)(pred_scores + (size_t)BB * MAX_DET); // B*300

  size_t lds1 = (size_t)NN * 16 + 512;   // 320,512 B of the 327,680 B WGP LDS
  fd_nms_per_class<<<BB * CC, THREADS, lds1, stream>>>(boxes, cls, cand_s, cand_i);

  size_t lds2 = (size_t)CM * 4 + 512;    // 96,512 B
  fd_topk_finalize<<<BB, THREADS, lds2, stream>>>(boxes, rel, cand_s, cand_i,
      out_boxes, out_scores, out_labels, pred_scores, pred_labels);
}